// ChaAttBlock_46763603919173
// MI455X (gfx1250) — compile-verified
//
#include <hip/hip_runtime.h>

// CDNA5 (gfx1250) wave32 fp32 WMMA implementation of channel attention:
//   energy = v v^T ; w = softmax(rowmax(energy) - energy) ; out = alpha*(w v) + x
typedef __attribute__((ext_vector_type(2))) float v2f;
typedef __attribute__((ext_vector_type(8))) float v8f;

#define BATCH 8
#define CH    256
#define SPA   16384  // 128*128

static __device__ __forceinline__ v8f wmma_f32(v2f a, v2f b, v8f c) {
  // D = A(16x4) * B(4x16) + C(16x16), fp32, wave32
  return __builtin_amdgcn_wmma_f32_16x16x4_f32(false, a, false, b, (short)0, c,
                                               false, false);
}

// ---------------------------------------------------------------------------
// Kernel 1: energy[b,c,d] = sum_s v[b,c,s] * v[b,d,s]
// One wave computes a 32x32 tile (2x2 grid of 16x16 WMMA tiles).
// A-frag and B-frag per-lane addressing are identical (contiguous float2).
// ---------------------------------------------------------------------------
__global__ __launch_bounds__(128) void energy_gemm(const float* __restrict__ x,
                                                   float* __restrict__ energy) {
  const int wave = blockIdx.x * (blockDim.x >> 5) + (threadIdx.x >> 5);
  const int lane = threadIdx.x & 31;
  const int b    = wave >> 6;        // 64 tiles (8x8 of 32x32) per batch
  const int t    = wave & 63;
  const int c0   = (t >> 3) << 5;
  const int d0   = (t & 7) << 5;

  const int row  = lane & 15;
  const int koff = (lane >> 4) << 1; // lanes 16..31 hold K=2,3

  const float* vb  = x + (size_t)b * CH * SPA;
  const float* pa0 = vb + (size_t)(c0 + row) * SPA + koff;
  const float* pa1 = vb + (size_t)(c0 + 16 + row) * SPA + koff;
  const float* pb0 = vb + (size_t)(d0 + row) * SPA + koff;
  const float* pb1 = vb + (size_t)(d0 + 16 + row) * SPA + koff;

  v8f acc00 = {}, acc01 = {}, acc10 = {}, acc11 = {};

  for (int s = 0; s < SPA; s += 4) {
    v2f a0 = *(const v2f*)(pa0 + s);
    v2f a1 = *(const v2f*)(pa1 + s);
    v2f b0 = *(const v2f*)(pb0 + s);
    v2f b1 = *(const v2f*)(pb1 + s);
    acc00 = wmma_f32(a0, b0, acc00);
    acc01 = wmma_f32(a0, b1, acc01);
    acc10 = wmma_f32(a1, b0, acc10);
    acc11 = wmma_f32(a1, b1, acc11);
  }

  float* eb = energy + (size_t)b * CH * CH;
  const int n   = lane & 15;
  const int mhi = (lane >> 4) << 3;  // upper half-lanes hold M+8
#pragma unroll
  for (int r = 0; r < 8; ++r) {
    const int m = mhi + r;
    eb[(size_t)(c0 + m)      * CH + (d0 + n)]      = acc00[r];
    eb[(size_t)(c0 + m)      * CH + (d0 + 16 + n)] = acc01[r];
    eb[(size_t)(c0 + 16 + m) * CH + (d0 + n)]      = acc10[r];
    eb[(size_t)(c0 + 16 + m) * CH + (d0 + 16 + n)] = acc11[r];
  }
}

// ---------------------------------------------------------------------------
// Kernel 2: in-place w = softmax(rowmax - e).
// Shift-invariance: softmax(max - e) == exp(min - e) / sum(exp(min - e)),
// which is numerically safe (exponents <= 0). One 256-thread block per row.
// ---------------------------------------------------------------------------
__global__ __launch_bounds__(256) void maxsub_softmax(float* __restrict__ e) {
  __shared__ float red[256];
  const int tid     = threadIdx.x;
  const size_t base = (size_t)blockIdx.x * CH;
  const float val   = e[base + tid];

  red[tid] = val;
  __syncthreads();
#pragma unroll
  for (int off = 128; off > 0; off >>= 1) {
    if (tid < off) red[tid] = fminf(red[tid], red[tid + off]);
    __syncthreads();
  }
  const float mn = red[0];
  __syncthreads();

  const float t = __expf(mn - val);
  red[tid] = t;
  __syncthreads();
#pragma unroll
  for (int off = 128; off > 0; off >>= 1) {
    if (tid < off) red[tid] += red[tid + off];
    __syncthreads();
  }
  e[base + tid] = t / red[0];
}

// ---------------------------------------------------------------------------
// Kernel 3: out = alpha * (w @ v) + x.
// One wave computes a 16x64 strip (1x4 N-blocking to reuse the w fragment).
// ---------------------------------------------------------------------------
__global__ __launch_bounds__(256) void av_gemm(const float* __restrict__ x,
                                               const float* __restrict__ w,
                                               const float* __restrict__ alpha,
                                               float* __restrict__ out) {
  const int wave = blockIdx.x * (blockDim.x >> 5) + (threadIdx.x >> 5);
  const int lane = threadIdx.x & 31;
  const int b    = wave >> 12;          // 4096 waves per batch
  const int rem  = wave & 4095;
  const int c0   = (rem >> 8) << 4;     // 16 M-tiles
  const int s0   = (rem & 255) << 6;    // 256 N-groups of width 64

  const int row  = lane & 15;
  const int koff = (lane >> 4) << 1;
  const int n    = lane & 15;

  const float* vb = x + (size_t)b * CH * SPA;
  const float* wb = w + (size_t)b * CH * CH;
  const float* pa = wb + (size_t)(c0 + row) * CH + koff;

  v8f acc0 = {}, acc1 = {}, acc2 = {}, acc3 = {};

  for (int k0 = 0; k0 < CH; k0 += 4) {
    v2f a = *(const v2f*)(pa + k0);
    const float* pb = vb + (size_t)(k0 + koff) * SPA + s0 + n;
    v2f b0, b1, b2, b3;
    b0.x = pb[0];  b0.y = pb[SPA];
    b1.x = pb[16]; b1.y = pb[SPA + 16];
    b2.x = pb[32]; b2.y = pb[SPA + 32];
    b3.x = pb[48]; b3.y = pb[SPA + 48];
    acc0 = wmma_f32(a, b0, acc0);
    acc1 = wmma_f32(a, b1, acc1);
    acc2 = wmma_f32(a, b2, acc2);
    acc3 = wmma_f32(a, b3, acc3);
  }

  const float al  = alpha[0];
  float* ob       = out + (size_t)b * CH * SPA;
  const int mhi   = (lane >> 4) << 3;
#pragma unroll
  for (int r = 0; r < 8; ++r) {
    const int m = c0 + mhi + r;
    const size_t base = (size_t)m * SPA + s0 + n;
    ob[base]      = al * acc0[r] + vb[base];
    ob[base + 16] = al * acc1[r] + vb[base + 16];
    ob[base + 32] = al * acc2[r] + vb[base + 32];
    ob[base + 48] = al * acc3[r] + vb[base + 48];
  }
}

// ---------------------------------------------------------------------------
extern "C" void kernel_launch(void* const* d_in, const int* in_sizes, int n_in,
                              void* d_out, int out_size, void* d_ws, size_t ws_size,
                              hipStream_t stream) {
  const float* x     = (const float*)d_in[0];
  const float* alpha = (const float*)d_in[1];
  float* out         = (float*)d_out;
  float* energy      = (float*)d_ws;   // B*C*C floats = 2 MB; reused as w

  // GEMM1: 8 batches * 64 tiles = 512 waves, 4 waves per block
  energy_gemm<<<BATCH * 64 / 4, 128, 0, stream>>>(x, energy);

  // softmax: one block per row, B*C = 2048 rows
  maxsub_softmax<<<BATCH * CH, 256, 0, stream>>>(energy);

  // GEMM2 + epilogue: 8 * 4096 waves, 8 waves per block
  av_gemm<<<BATCH * 4096 / 8, 256, 0, stream>>>(x, energy, alpha, out);
}